// deltaLayer_57294863728910
// MI455X (gfx1250) — compile-verified
//
#include <hip/hip_runtime.h>

// Cost-volume: out[b,c,i,j] = |l[b,c,j] - r[b,c,i]|, (2,128,1,900) -> (2,128,900,900).
// Store-bandwidth bound: 829 MB @ 23.3 TB/s ~= 36us floor. Each wave produces a 16x64
// strip via 4x V_WMMA_F32_16X16X4_F32 (rank-2 trick [1,-r_i] x [l_j,1], bit-exact fp32),
// wave-local LDS transpose (no block barrier; DS ops are in-order per wave), then
// unguarded 16B non-temporal b128 stores on the interior fast path.

typedef __attribute__((ext_vector_type(2))) float v2f;
typedef __attribute__((ext_vector_type(4))) float v4f;
typedef __attribute__((ext_vector_type(8))) float v8f;

#define W               900
#define TPD             57                    // ceil(900/16) column tiles
#define STRIPS          15                    // 14 strips of 4 tiles + 1 strip of 1
#define NSLICES         256                   // BS*CHAN
#define WAVES_TOTAL     (NSLICES * TPD * STRIPS)   // 218880
#define WAVES_PER_BLOCK 8
#define LDS_STRIDE      68                    // 64 cols + 4 pad: 16B-aligned, conflict-free writes

__global__ __launch_bounds__(256) void delta_cost_volume_wmma(
    const float* __restrict__ l, const float* __restrict__ r, float* __restrict__ out)
{
  __shared__ __align__(16) float lds[WAVES_PER_BLOCK][16 * LDS_STRIDE];

  const int lane = threadIdx.x & 31;
  const int wave = threadIdx.x >> 5;
  const unsigned w = blockIdx.x * WAVES_PER_BLOCK + wave;  // grid covers waves exactly

  const unsigned slice = w / (TPD * STRIPS);               // (b*CHAN + c)
  const unsigned rem   = w % (TPD * STRIPS);
  const int ti = rem / STRIPS;                 // output row tile (i <- r)
  const int s  = rem % STRIPS;                 // strip of column tiles (j <- l)
  const int nt   = (s == STRIPS - 1) ? 1 : 4;  // wave-uniform
  const int row0 = ti * 16;
  const int col0 = s * 64;                     // strip 14 -> col0 = 896 (tile tj=56)

  const float* lrow = l + (size_t)slice * W;
  const float* rrow = r + (size_t)slice * W;

  // A fragment (shared by all 4 WMMAs): col0 = 1, col1 = -r_i, K=2,3 zero (lanes 16..31).
  const int idx = lane & 15;
  int mi = row0 + idx; mi = (mi < W) ? mi : (W - 1);       // clamp; rows >= 900 discarded at store
  const float rv = rrow[mi];
  const bool lo = (lane < 16);
  v2f a;  a.x = lo ? 1.0f : 0.0f;  a.y = lo ? -rv : 0.0f;

  const int mbase = (lane >> 4) << 3;          // D: VGPR v -> row v + 8*(lane>=16), col lane&15
  float* t = &lds[wave][0];

#pragma unroll
  for (int k = 0; k < 4; ++k) {
    if (k < nt) {                              // wave-uniform guard: EXEC stays all-ones
      int nj = col0 + k * 16 + idx; nj = (nj < W) ? nj : (W - 1);
      const float lv = lrow[nj];
      v2f bm; bm.x = lo ? lv : 0.0f;  bm.y = lo ? 1.0f : 0.0f;
      v8f c = {};
      // D[i,j] = 1*l_j + (-r_i)*1 + 0 + 0  (exact fp32, order-independent)
      v8f d = __builtin_amdgcn_wmma_f32_16x16x4_f32(
          false, a, false, bm, (short)0, c, false, false);
#pragma unroll
      for (int v = 0; v < 8; ++v)
        t[(mbase + v) * LDS_STRIDE + k * 16 + idx] = __builtin_fabsf(d[v]);
    }
  }

  // Wave-local LDS RAW: DS instructions from one wave complete in order; only need to
  // stop the compiler from reordering. No s_barrier, no cross-wave coupling.
  __asm__ __volatile__("" ::: "memory");
  __builtin_amdgcn_wave_barrier();

  float* outSlice = out + (size_t)slice * W * W;

  if (nt == 4) {
    // 16 rows x 16 float4-groups; lane -> (tr0 = lane>>4, cg = lane&15), h steps tr by 2.
    const int tr0 = lane >> 4;
    const int cg  = idx;
    const float* lbase = &t[tr0 * LDS_STRIDE + cg * 4];
    float* obase = outSlice + (size_t)(row0 + tr0) * W + col0 + cg * 4;
    if (row0 + 16 <= W) {                      // interior: no guards at all
#pragma unroll
      for (int h = 0; h < 8; ++h) {
        const v4f val = *(const v4f*)(lbase + h * 2 * LDS_STRIDE);
        __builtin_nontemporal_store(val, (v4f*)(obase + (size_t)h * 2 * W));
      }
    } else {                                   // row-edge tile (ti == 56): rows 896..899 valid
#pragma unroll
      for (int h = 0; h < 8; ++h) {
        if (row0 + tr0 + 2 * h < W) {
          const v4f val = *(const v4f*)(lbase + h * 2 * LDS_STRIDE);
          __builtin_nontemporal_store(val, (v4f*)(obase + (size_t)h * 2 * W));
        }
      }
    }
  } else {
    // Single edge tile tj=56: cols 896..899 valid -> only colgroup 0 (900 % 4 == 0).
#pragma unroll
    for (int h = 0; h < 2; ++h) {
      const int tr = (lane >> 2) + 8 * h;
      const int cg = lane & 3;
      if (cg == 0 && row0 + tr < W) {
        const v4f val = *(const v4f*)&t[tr * LDS_STRIDE];
        __builtin_nontemporal_store(val, (v4f*)(outSlice + (size_t)(row0 + tr) * W + col0));
      }
    }
  }
}

extern "C" void kernel_launch(void* const* d_in, const int* in_sizes, int n_in,
                              void* d_out, int out_size, void* d_ws, size_t ws_size,
                              hipStream_t stream) {
  const float* l = (const float*)d_in[0];   // encoded_l (2,128,1,900) fp32
  const float* r = (const float*)d_in[1];   // encoded_r (2,128,1,900) fp32
  float* out = (float*)d_out;               // (2,128,900,900) fp32

  const unsigned nblocks = WAVES_TOTAL / WAVES_PER_BLOCK;  // 27360, exact
  delta_cost_volume_wmma<<<dim3(nblocks), dim3(256), 0, stream>>>(l, r, out);
}